// BOCPD_GPTS_18932215841132
// MI455X (gfx1250) — compile-verified
//
#include <hip/hip_runtime.h>
#include <hip/hip_bf16.h>
#include <math.h>

// BOCPD-GPTS on MI455X (gfx1250).
// Key insight: the run-length recursion telescopes (r stays normalized), so
// nlml = sum_t upm(t) and all 2048 GP predictions are independent.
// One 128-thread (4xwave32) workgroup per t does a blocked fp32 Cholesky of a
// 128x128 (masked) kernel matrix in LDS, with the SYRK trailing updates on
// V_WMMA_F32_16X16X4_F32, then a wave-level dual forward substitution.

typedef __attribute__((ext_vector_type(2))) float v2f;
typedef __attribute__((ext_vector_type(8))) float v8f;

#define N 128          // padded window (Wm=127 active rows + 1 identity row)
#define STR 132        // LDS row stride in floats (132 % 64 == 4 -> conflict-free tiles)
#define NB 16          // Cholesky block size (WMMA tile)
#define NBLK (N / NB)  // 8 block rows/cols

__global__ __launch_bounds__(128) void bocpd_upm_kernel(
    const float* __restrict__ xg, const float* __restrict__ yg,
    const float* __restrict__ ll, const float* __restrict__ lo,
    const float* __restrict__ ln, const int* __restrict__ wsz,
    float* __restrict__ upm_out, int T)
{
    extern __shared__ float smem[];
    float* Ks = smem;            // N*STR
    float* xs = smem + N * STR;  // N
    float* ys = xs + N;          // N
    float* b1 = ys + N;          // N  (kx, masked)
    float* b2 = b1 + N;          // N  (yw*m)

    const int tid  = threadIdx.x;
    const int wave = tid >> 5;
    const int lane = tid & 31;
    const int t    = blockIdx.x + 1;          // t = 1..T

    const int W  = wsz[0];                    // 128
    const int Wm = W - 1;                     // 127
    int s = t - W; if (s < 0) s = 0;
    int Leff = t - 1; if (Leff > Wm) Leff = Wm; if (Leff < 1) Leff = 1;

    const float ell2  = __expf(2.0f * ll[0]);
    const float sig2  = __expf(lo[0]);
    const float noise = __expf(ln[0]);
    const float inv2e = 0.5f / ell2;
    const float xstar = xg[t - 1];
    const float ystar = yg[t - 1];

    // ---- load window + RHS vectors -------------------------------------
    {
        float xv = 0.0f, yv = 0.0f;
        if (tid < Wm) { xv = xg[s + tid]; yv = yg[s + tid]; }
        xs[tid] = xv; ys[tid] = yv;
        float mi = (tid < Leff) ? 1.0f : 0.0f;
        float dx = xv - xstar;
        b1[tid] = mi * sig2 * __expf(-dx * dx * inv2e);
        b2[tid] = mi * yv;
    }
    __syncthreads();

    // ---- build masked RBF kernel matrix K (full symmetric) -------------
    for (int idx = tid; idx < N * N; idx += 128) {
        int i = idx >> 7, j = idx & (N - 1);
        float mi = (i < Leff) ? 1.0f : 0.0f;
        float mj = (j < Leff) ? 1.0f : 0.0f;
        float d  = xs[i] - xs[j];
        float v  = sig2 * __expf(-d * d * inv2e) * mi * mj;
        if (i == j) v += noise * mi + (1.0f - mi);
        Ks[i * STR + j] = v;
    }
    __syncthreads();

    // ---- blocked right-looking Cholesky (lower) ------------------------
    for (int k = 0; k < NBLK; ++k) {
        const int k16 = k * NB;

        // (a) factor 16x16 diagonal block (scalar, column-by-column)
        for (int c = 0; c < NB; ++c) {
            if (tid == 0)
                Ks[(k16 + c) * STR + k16 + c] = sqrtf(Ks[(k16 + c) * STR + k16 + c]);
            __syncthreads();
            if (tid > c && tid < NB)
                Ks[(k16 + tid) * STR + k16 + c] /= Ks[(k16 + c) * STR + k16 + c];
            __syncthreads();
            if (tid > c && tid < NB) {
                float lic = Ks[(k16 + tid) * STR + k16 + c];
                for (int j2 = c + 1; j2 <= tid; ++j2)
                    Ks[(k16 + tid) * STR + k16 + j2] -= lic * Ks[(k16 + j2) * STR + k16 + c];
            }
            __syncthreads();
        }

        // (b) panel triangular solve: rows below diag block, one row/thread
        const int nrows = (NBLK - 1 - k) * NB;
        if (tid < nrows) {
            const int row = (k + 1) * NB + tid;
            float a[NB];
            #pragma unroll
            for (int c = 0; c < NB; ++c) a[c] = Ks[row * STR + k16 + c];
            #pragma unroll
            for (int c = 0; c < NB; ++c) {
                float acc = a[c];
                #pragma unroll
                for (int c2 = 0; c2 < NB; ++c2)
                    if (c2 < c) acc -= a[c2] * Ks[(k16 + c) * STR + k16 + c2];
                a[c] = acc / Ks[(k16 + c) * STR + k16 + c];
            }
            #pragma unroll
            for (int c = 0; c < NB; ++c) Ks[row * STR + k16 + c] = a[c];
        }
        __syncthreads();

        // (c) trailing SYRK update on fp32 WMMA: C(i,j) -= L(i,k) * L(j,k)^T
        const int nb = NBLK - 1 - k;
        const int ntiles = nb * (nb + 1) / 2;
        for (int tIdx = wave; tIdx < ntiles; tIdx += 4) {   // wave-uniform: EXEC all-1s
            int bi = 0;
            while ((bi + 1) * (bi + 2) / 2 <= tIdx) bi++;
            int bj = tIdx - bi * (bi + 1) / 2;
            const int I = (k + 1 + bi) * NB;
            const int J = (k + 1 + bj) * NB;
            const int half = lane >> 4, l = lane & 15;

            v8f c;
            #pragma unroll
            for (int v = 0; v < 8; ++v) c[v] = Ks[(I + v + 8 * half) * STR + J + l];

            #pragma unroll
            for (int ch = 0; ch < 4; ++ch) {
                // A (16x4): A[m][kk] = -L_ik[m][kk]; lane: m=l, kk = 4ch + 2*half + v
                v2f a, b;
                a[0] = -Ks[(I + l) * STR + k16 + 4 * ch + 2 * half + 0];
                a[1] = -Ks[(I + l) * STR + k16 + 4 * ch + 2 * half + 1];
                // B (4x16): B[kk][n] = L_jk[n][kk]; lane: n=l
                b[0] =  Ks[(J + l) * STR + k16 + 4 * ch + 2 * half + 0];
                b[1] =  Ks[(J + l) * STR + k16 + 4 * ch + 2 * half + 1];
                c = __builtin_amdgcn_wmma_f32_16x16x4_f32(
                        /*neg_a=*/false, a, /*neg_b=*/false, b,
                        /*c_mod=*/(short)0, c, /*reuse_a=*/false, /*reuse_b=*/false);
            }

            #pragma unroll
            for (int v = 0; v < 8; ++v) Ks[(I + v + 8 * half) * STR + J + l] = c[v];
        }
        __syncthreads();
    }

    // ---- dual forward substitution (wave 0), fused dot products --------
    if (wave == 0) {
        float b1r[4], b2r[4];
        #pragma unroll
        for (int q = 0; q < 4; ++q) {
            b1r[q] = b1[lane + 32 * q];
            b2r[q] = b2[lane + 32 * q];
        }
        float accAV = 0.0f, accAA = 0.0f;
        for (int j = 0; j < N; ++j) {
            const int slot = j >> 5, owner = j & 31;
            float c1 = (slot == 0) ? b1r[0] : (slot == 1) ? b1r[1] : (slot == 2) ? b1r[2] : b1r[3];
            float c2 = (slot == 0) ? b2r[0] : (slot == 1) ? b2r[1] : (slot == 2) ? b2r[2] : b2r[3];
            const float Ljj = Ks[j * STR + j];
            const float s1 = __shfl(c1, owner, 32) / Ljj;
            const float s2 = __shfl(c2, owner, 32) / Ljj;
            accAV += s1 * s2;
            accAA += s1 * s1;
            #pragma unroll
            for (int q = 0; q < 4; ++q) {
                const int r = lane + 32 * q;
                if (r > j) {
                    const float lij = Ks[r * STR + j];
                    b1r[q] -= lij * s1;
                    b2r[q] -= lij * s2;
                }
            }
        }
        if (lane == 0) {
            const float mu  = accAV;
            const float var = sig2 - accAA;
            const float d   = ystar - mu;
            const float upm = __expf(-0.5f * (__logf(6.2831853071795864f * var) + d * d / var));
            upm_out[t - 1] = upm;
        }
    }
}

// Deterministic sum reduction of the 2048 upm values -> nlml scalar.
__global__ __launch_bounds__(256) void bocpd_reduce_kernel(
    const float* __restrict__ upm, float* __restrict__ out, int n)
{
    __shared__ float sd[256];
    float acc = 0.0f;
    for (int i = threadIdx.x; i < n; i += 256) acc += upm[i];
    sd[threadIdx.x] = acc;
    __syncthreads();
    for (int off = 128; off > 0; off >>= 1) {
        if ((int)threadIdx.x < off) sd[threadIdx.x] += sd[threadIdx.x + off];
        __syncthreads();
    }
    if (threadIdx.x == 0) out[0] = sd[0];
}

extern "C" void kernel_launch(void* const* d_in, const int* in_sizes, int n_in,
                              void* d_out, int out_size, void* d_ws, size_t ws_size,
                              hipStream_t stream) {
    const float* X   = (const float*)d_in[0];
    const float* Y   = (const float*)d_in[1];
    const float* ll  = (const float*)d_in[2];   // log_lengthscale
    const float* lo  = (const float*)d_in[3];   // log_outputscale
    const float* ln  = (const float*)d_in[4];   // log_noise
    // d_in[5] = hazard: algebraically cancels out of nlml (r stays normalized)
    const int*   wsz = (const int*)d_in[6];     // window_size

    const int T = in_sizes[0];                  // 2048
    float* upm = (float*)d_ws;                  // T floats of scratch

    const size_t lds_bytes = (size_t)(N * STR + 4 * N) * sizeof(float);
    bocpd_upm_kernel<<<T, 128, lds_bytes, stream>>>(X, Y, ll, lo, ln, wsz, upm, T);
    bocpd_reduce_kernel<<<1, 256, 0, stream>>>(upm, (float*)d_out, T);
}